// GlandClassificationHead_78271484002506
// MI455X (gfx1250) — compile-verified
//
#include <hip/hip_runtime.h>
#include <hip/hip_bf16.h>

// GlandClassificationHead for MI455X (gfx1250, wave32).
//
// Phase 1 (bandwidth-bound): masked class-sum pooling over features
//   [B=8,C=512,H=128,W=128] fp32 = 256 MiB, read exactly once -> ~11us floor
//   at 23.3 TB/s. Streaming select-accumulate with b128 (float4)
//   non-temporal loads so the kernel is vmem-issue-light and HBM-bound;
//   the tiny mask (512 KB) stays L2-resident for its 64x reuse.
// Phase 2 (tiny): 24x512 @ 512x256 -> relu -> @ 256x4 MLP, done with
//   V_WMMA_F32_16X16X4_F32 (fp32 WMMA, matches reference precision).

typedef __attribute__((ext_vector_type(2))) float v2f;
typedef __attribute__((ext_vector_type(4))) float v4f;
typedef __attribute__((ext_vector_type(8))) float v8f;
typedef __attribute__((ext_vector_type(4))) int   v4i;

#define B_DIM 8
#define C_DIM 512
#define HW_DIM 16384       // 128*128
#define HW_V4 (HW_DIM / 4) // 4096 float4 per plane
#define K_CLS 3
#define CPB 8              // channels per pooling block
#define CH_DIM 256         // C/2
#define H_STRIDE 260       // padded LDS stride for h
#define MIN_GLAND 4.0f

// ---------------------------------------------------------------------------
// Kernel 1: per-(b, channel-chunk) masked sums. One block fully reduces its
// 8 channels x 3 classes -> direct global writes, no atomics needed.
// b128 loads: each thread owns 4 consecutive pixels per iteration.
// ---------------------------------------------------------------------------
__global__ void gland_pool_kernel(const float* __restrict__ feat,
                                  const int* __restrict__ mask,
                                  float* __restrict__ sums) {
    const int b   = blockIdx.y;
    const int c0  = blockIdx.x * CPB;
    const int tid = threadIdx.x;

    float acc[CPB][K_CLS];
#pragma unroll
    for (int i = 0; i < CPB; ++i)
#pragma unroll
        for (int k = 0; k < K_CLS; ++k) acc[i][k] = 0.0f;

    const v4f* fb = (const v4f*)(feat + ((size_t)b * C_DIM + c0) * HW_DIM);
    const v4i* mb = (const v4i*)(mask + (size_t)b * HW_DIM);

    for (int p = tid; p < HW_V4; p += 256) {
        const v4i m = mb[p];                       // b128, L2-resident, 64x reuse
        float oh[4][K_CLS];
#pragma unroll
        for (int j = 0; j < 4; ++j) {
            oh[j][0] = (m[j] == 1) ? 1.0f : 0.0f;
            oh[j][1] = (m[j] == 2) ? 1.0f : 0.0f;
            oh[j][2] = (m[j] == 3) ? 1.0f : 0.0f;
        }
#pragma unroll
        for (int i = 0; i < CPB; ++i) {
            // streaming b128: single-use data, non-temporal (TH=NT)
            const v4f f = __builtin_nontemporal_load(&fb[(size_t)i * HW_V4 + p]);
#pragma unroll
            for (int k = 0; k < K_CLS; ++k) {
                acc[i][k] = fmaf(f[0], oh[0][k], acc[i][k]);
                acc[i][k] = fmaf(f[1], oh[1][k], acc[i][k]);
                acc[i][k] = fmaf(f[2], oh[2][k], acc[i][k]);
                acc[i][k] = fmaf(f[3], oh[3][k], acc[i][k]);
            }
        }
    }

    // wave32 reduction
#pragma unroll
    for (int i = 0; i < CPB; ++i)
#pragma unroll
        for (int k = 0; k < K_CLS; ++k)
#pragma unroll
            for (int off = 16; off > 0; off >>= 1)
                acc[i][k] += __shfl_xor(acc[i][k], off, 32);

    __shared__ float red[8][CPB * K_CLS];
    const int wave = tid >> 5;
    const int lane = tid & 31;
    if (lane == 0) {
#pragma unroll
        for (int i = 0; i < CPB; ++i)
#pragma unroll
            for (int k = 0; k < K_CLS; ++k)
                red[wave][i * K_CLS + k] = acc[i][k];
    }
    __syncthreads();

    if (tid < CPB * K_CLS) {
        float s = 0.0f;
#pragma unroll
        for (int w = 0; w < 8; ++w) s += red[w][tid];
        const int i = tid / K_CLS;
        const int k = tid % K_CLS;
        // sums layout: row = b*3 + k (24 rows used, 32 allocated), col = c
        sums[((size_t)b * K_CLS + k) * C_DIM + (c0 + i)] = s;
    }
}

// ---------------------------------------------------------------------------
// Kernel 2: per-batch class counts + zero-fill padded sum rows 24..31.
// ---------------------------------------------------------------------------
__global__ void gland_count_kernel(const int* __restrict__ mask,
                                   float* __restrict__ counts,
                                   float* __restrict__ sums) {
    const int b   = blockIdx.x;
    const int tid = threadIdx.x;

    int c[K_CLS] = {0, 0, 0};
    const v4i* mb = (const v4i*)(mask + (size_t)b * HW_DIM);
    for (int p = tid; p < HW_V4; p += 256) {
        const v4i m = mb[p];
#pragma unroll
        for (int j = 0; j < 4; ++j) {
            c[0] += (m[j] == 1);
            c[1] += (m[j] == 2);
            c[2] += (m[j] == 3);
        }
    }
#pragma unroll
    for (int k = 0; k < K_CLS; ++k)
#pragma unroll
        for (int off = 16; off > 0; off >>= 1)
            c[k] += __shfl_xor(c[k], off, 32);

    __shared__ int red[8][K_CLS];
    const int wave = tid >> 5;
    const int lane = tid & 31;
    if (lane == 0) {
#pragma unroll
        for (int k = 0; k < K_CLS; ++k) red[wave][k] = c[k];
    }
    __syncthreads();

    if (tid < K_CLS) {
        int s = 0;
#pragma unroll
        for (int w = 0; w < 8; ++w) s += red[w][tid];
        counts[b * K_CLS + tid] = (float)s;
    }
    // zero-fill M-padding row (24+b) of sums so WMMA reads defined zeros
    for (int j = tid; j < C_DIM; j += 256)
        sums[(size_t)(24 + b) * C_DIM + j] = 0.0f;
}

// ---------------------------------------------------------------------------
// Kernel 3: means -> WMMA fp32 layer1 -> relu -> scalar layer2 -> mask/valid.
// One block, 1024 threads = 32 wave32 waves; wave w owns h-tile
// (m_tile = w>>4, n_tile = w&15). K loop: 512 in steps of 4 per
// V_WMMA_F32_16X16X4_F32.
//
// Fragment layouts (ISA 7.12.2, 32-bit):
//   A 16x4: V0 lanes0-15:(M=lane,K=0) lanes16-31:(M=lane-16,K=2); V1: K=1/K=3
//   B 4x16: V0 lanes0-15:(K=0,N=lane) lanes16-31:(K=2,N=lane-16); V1: K=1/K=3
//   C/D 16x16: VGPR r: lanes0-15 M=r, lanes16-31 M=r+8
// ---------------------------------------------------------------------------
__global__ void gland_mlp_kernel(const float* __restrict__ sums,
                                 const float* __restrict__ counts,
                                 const float* __restrict__ W1,
                                 const float* __restrict__ b1,
                                 const float* __restrict__ W2,
                                 const float* __restrict__ b2,
                                 float* __restrict__ out) {
    __shared__ float lds_h[32 * H_STRIDE];
    __shared__ float inv_cnt[32];

    const int tid = threadIdx.x;
    if (tid < 32) {
        const float c = (tid < 24) ? counts[tid] : 1.0f;
        inv_cnt[tid] = 1.0f / fmaxf(c, 1.0f);
    }
    __syncthreads();

    const int wave   = tid >> 5;     // 0..31
    const int lane   = tid & 31;
    const int m_tile = wave >> 4;    // 0..1
    const int n_tile = wave & 15;    // 0..15
    const int half   = lane >> 4;    // 0: K+0/K+1, 1: K+2/K+3
    const int lmod   = lane & 15;

    const int arow = m_tile * 16 + lmod;   // A-matrix row for this lane
    const int bcol = n_tile * 16 + lmod;   // B-matrix col for this lane
    const float ainv = inv_cnt[arow];
    const int ka_off = half * 2;

    v8f acc = {};
    const float* srow = sums + (size_t)arow * C_DIM;
    for (int kk = 0; kk < C_DIM; kk += 4) {
        const int ka = kk + ka_off;
        v2f a, bf;
        a.x  = srow[ka]     * ainv;              // mean = sum / max(count,1)
        a.y  = srow[ka + 1] * ainv;
        bf.x = W1[(size_t)ka * CH_DIM + bcol];
        bf.y = W1[(size_t)(ka + 1) * CH_DIM + bcol];
        // 8 args: (neg_a, A, neg_b, B, c_mod, C, reuse_a, reuse_b)
        acc = __builtin_amdgcn_wmma_f32_16x16x4_f32(
            false, a, false, bf, (short)0, acc, false, false);
    }

    const float bias = b1[bcol];
#pragma unroll
    for (int r = 0; r < 8; ++r) {
        const int mrow = m_tile * 16 + half * 8 + r;
        lds_h[mrow * H_STRIDE + bcol] = fmaxf(acc[r] + bias, 0.0f);
    }
    __syncthreads();

    // Layer 2: [24,256] @ [256,4] + b2, validity masking, valid flags.
    if (tid < 24 * 4) {
        const int row = tid >> 2;  // 0..23  (= b*3 + k)
        const int col = tid & 3;
        float s = b2[col];
        const float* hrow = &lds_h[row * H_STRIDE];
        for (int j = 0; j < CH_DIM; ++j)
            s += hrow[j] * W2[j * 4 + col];
        const bool valid = counts[row] > MIN_GLAND;
        out[row * 4 + col] = valid ? s : 0.0f;
        if (col == 0) out[96 + row] = valid ? 1.0f : 0.0f;
    }
}

// ---------------------------------------------------------------------------
extern "C" void kernel_launch(void* const* d_in, const int* in_sizes, int n_in,
                              void* d_out, int out_size, void* d_ws, size_t ws_size,
                              hipStream_t stream) {
    const float* features = (const float*)d_in[0];  // [8,512,128,128]
    const int*   seg_mask = (const int*)d_in[1];    // [8,128,128]
    const float* W1       = (const float*)d_in[2];  // [512,256]
    const float* b1       = (const float*)d_in[3];  // [256]
    const float* W2       = (const float*)d_in[4];  // [256,4]
    const float* b2       = (const float*)d_in[5];  // [4]
    float* out = (float*)d_out;                     // 96 logits + 24 valid

    // workspace: sums[32 rows x 512] then counts[24]
    float* ws_sums   = (float*)d_ws;
    float* ws_counts = ws_sums + 32 * C_DIM;

    dim3 pool_grid(C_DIM / CPB, B_DIM);  // 64 x 8 blocks
    gland_pool_kernel<<<pool_grid, 256, 0, stream>>>(features, seg_mask, ws_sums);
    gland_count_kernel<<<B_DIM, 256, 0, stream>>>(seg_mask, ws_counts, ws_sums);
    gland_mlp_kernel<<<1, 1024, 0, stream>>>(ws_sums, ws_counts, W1, b1, W2, b2, out);
}